// SegWrapper_48241072668963
// MI455X (gfx1250) — compile-verified
//
#include <hip/hip_runtime.h>

#define BS     2
#define NF     16
#define CH     32
#define HW     16384      // 128*128
#define NMASK  8
#define NCLS   3
#define MID    (NF/2)

#define TILE   128        // hw positions per LDS tile (per buffer)
#define CHUNK  2048       // hw positions per block
#define NT     (CHUNK/TILE)   // 16 tiles, double buffered
#define FSTR   132        // 128 + 4 pad (132 % 64 == 4 -> conflict-free)
#define MSTR   132

typedef __attribute__((ext_vector_type(2))) float v2f;
typedef __attribute__((ext_vector_type(8))) float v8f;
typedef __attribute__((ext_vector_type(4))) unsigned int u32x4;
typedef __attribute__((ext_vector_type(8))) int i32x8;
typedef __attribute__((ext_vector_type(4))) int i32x4;

// ---------------------------------------------------------------------------
// TDM: 2-D tile load global->LDS. rows x TILE dwords, row stride in dwords,
// LDS rows padded by +4 dwords via pad_interval/pad_amount.
// D# layout per CDNA5 ISA 8.3/8.4. Groups 2/3 zero (<=2D tensor).
// pad_interval code 6 => pad every 128 DWORDs; pad_amount code 3 => 4 DWORDs.
// ---------------------------------------------------------------------------
__device__ __forceinline__ void tdm_load_tile(unsigned lds_byte_off,
                                              const float* gaddr,
                                              unsigned rows,
                                              unsigned tensor_cols,
                                              unsigned row_stride_dw) {
    unsigned long long ga = (unsigned long long)(size_t)gaddr;
    u32x4 g0;
    g0[0] = 1u;                                   // count=1, user descriptor
    g0[1] = lds_byte_off;                         // lds_addr [63:32]
    g0[2] = (unsigned)(ga & 0xFFFFFFFFu);         // global_addr [95:64]
    g0[3] = (unsigned)((ga >> 32) & 0x01FFFFFFu)  // global_addr [120:96]
          | (2u << 30);                           // type=2 ("image") [127:126]
    i32x8 g1;
    g1[0] = (int)((2u << 16)      // data_size = 4B
                | (1u << 20)      // pad_enable
                | (6u << 22)      // pad_interval: 128 DWORDs
                | (3u << 25));    // pad_amount: 4 DWORDs
    g1[1] = (int)((tensor_cols & 0xFFFFu) << 16);             // tensor_dim0[15:0]
    g1[2] = (int)(((tensor_cols >> 16) & 0xFFFFu)             // tensor_dim0[31:16]
                | ((rows & 0xFFFFu) << 16));                  // tensor_dim1[15:0]
    g1[3] = (int)(((unsigned)TILE & 0xFFFFu) << 16);          // tile_dim0 = 128
    g1[4] = (int)(rows & 0xFFFFu);                            // tile_dim1 (tile_dim2=0)
    g1[5] = (int)row_stride_dw;                               // tensor_dim0_stride[31:0]
    g1[6] = 0;
    g1[7] = 0;
    i32x4 z4 = {};
    i32x8 z8 = {};
    __builtin_amdgcn_tensor_load_to_lds(g0, g1, z4, z4, z8, 0);
}

// ---------------------------------------------------------------------------
__global__ void zero_ws_kernel(float* __restrict__ ws) {
    ws[threadIdx.x] = 0.0f;
}

// ---------------------------------------------------------------------------
// masks[b,m,hw] = sum_c dec[b,MID,c,hw]*seg_w[m,c] + seg_b[m]
// ---------------------------------------------------------------------------
__global__ void masks_kernel(const float* __restrict__ dec,
                             const float* __restrict__ seg_w,
                             const float* __restrict__ seg_b,
                             float* __restrict__ masks) {
    __shared__ float w[NMASK * CH];
    int tid = threadIdx.x;
    w[tid] = seg_w[tid];
    __syncthreads();

    int b  = blockIdx.y;
    int hw = blockIdx.x * 256 + tid;
    const float* p = dec + ((size_t)(b * NF + MID)) * CH * HW + hw;

    float acc[NMASK];
    #pragma unroll
    for (int m = 0; m < NMASK; ++m) acc[m] = seg_b[m];

    #pragma unroll 4
    for (int c = 0; c < CH; ++c) {
        float f = p[(size_t)c * HW];
        #pragma unroll
        for (int m = 0; m < NMASK; ++m)
            acc[m] = fmaf(f, w[m * CH + c], acc[m]);
    }
    #pragma unroll
    for (int m = 0; m < NMASK; ++m)
        masks[((size_t)(b * NMASK + m)) * HW + hw] = acc[m];
}

// ---------------------------------------------------------------------------
// pooled[b,m,c] += sum_{f,hw} dec[b,f,c,hw] * masks[b,m,hw]
// TDM double-buffered LDS tiles + V_WMMA_F32_16X16X4_F32.
// Grid (BS*NF, HW/CHUNK), block 256 = 8 waves; wave w owns K slice
// [w*16, w*16+16) of each 128-wide tile.
// ---------------------------------------------------------------------------
__global__ void pooled_kernel(const float* __restrict__ dec,
                              const float* __restrict__ masks,
                              float* __restrict__ ws_pooled) {
    __shared__ float ldsF[2][CH * FSTR];     // 2 x 32 rows x 128 (+4 pad)
    __shared__ float ldsM[2][16 * MSTR];     // 2 x (8 mask rows + 8 zero rows)
    __shared__ float ldsAcc[CH * NMASK];

    int tid    = threadIdx.x;
    int wid    = tid >> 5;
    int lane   = tid & 31;
    int lane15 = lane & 15;
    int hi     = lane >> 4;          // 32-bit WMMA: hi lanes take K+2/K+3
    int koff   = hi << 1;

    int bf = blockIdx.x;
    int b  = bf / NF;
    int hwbase = blockIdx.y * CHUNK;
    const float* gF = dec   + (size_t)bf * CH * HW;
    const float* gM = masks + (size_t)b * NMASK * HW;

    // zero reduction buffer + N-padding mask rows (both buffers); these LDS
    // regions are disjoint from everything TDM writes.
    ldsAcc[tid] = 0.0f;
    for (int i = tid; i < 8 * MSTR; i += 256) {
        ldsM[0][8 * MSTR + i] = 0.0f;
        ldsM[1][8 * MSTR + i] = 0.0f;
    }

    // prologue: DMA tile 0 into buffer 0 (wave 0 issues; EXEC-independent op)
    if (wid == 0) {
        tdm_load_tile((unsigned)(size_t)&ldsF[0][0], gF + hwbase, CH, HW, HW);
        tdm_load_tile((unsigned)(size_t)&ldsM[0][0], gM + hwbase, NMASK, HW, HW);
    }

    v8f acc0 = {};   // channels 0..15
    v8f acc1 = {};   // channels 16..31

    for (int t = 0; t < NT; ++t) {
        if (wid == 0) {
            if (t + 1 < NT) {
                int hwn = hwbase + (t + 1) * TILE;
                int nb  = (t + 1) & 1;
                tdm_load_tile((unsigned)(size_t)&ldsF[nb][0], gF + hwn, CH, HW, HW);
                tdm_load_tile((unsigned)(size_t)&ldsM[nb][0], gM + hwn, NMASK, HW, HW);
                __builtin_amdgcn_s_wait_tensorcnt(2);  // tile t done, t+1 in flight
            } else {
                __builtin_amdgcn_s_wait_tensorcnt(0);
            }
        }
        __syncthreads();   // releases all waves once tile t is in LDS

        const float* F  = ldsF[t & 1];
        const float* Mm = ldsM[t & 1];
        #pragma unroll
        for (int ks = 0; ks < 4; ++ks) {
            int kb = (wid << 4) + (ks << 2) + koff;
            v2f bb = *(const v2f*)&Mm[lane15 * MSTR + kb];
            v2f a0 = *(const v2f*)&F[lane15 * FSTR + kb];
            v2f a1 = *(const v2f*)&F[(16 + lane15) * FSTR + kb];
            acc0 = __builtin_amdgcn_wmma_f32_16x16x4_f32(
                       false, a0, false, bb, (short)0, acc0, false, false);
            acc1 = __builtin_amdgcn_wmma_f32_16x16x4_f32(
                       false, a1, false, bb, (short)0, acc1, false, false);
        }
        __syncthreads();   // tile t fully consumed before buffer reuse
    }

    // reduce 8 waves' D tiles via LDS float atomics.
    // D layout: VGPR r, lanes 0-15 -> (M=r, N=lane); lanes 16-31 -> (M=r+8).
    int n = lane15;
    if (n < NMASK) {
        int cbase = hi << 3;
        #pragma unroll
        for (int r = 0; r < 8; ++r) {
            atomicAdd(&ldsAcc[(cbase + r)      * NMASK + n], acc0[r]);
            atomicAdd(&ldsAcc[(16 + cbase + r) * NMASK + n], acc1[r]);
        }
    }
    __syncthreads();

    int c = tid >> 3, mm = tid & 7;
    atomicAdd(&ws_pooled[b * (NMASK * CH) + mm * CH + c], ldsAcc[c * NMASK + mm]);
}

// ---------------------------------------------------------------------------
__global__ void logits_kernel(const float* __restrict__ ws_pooled,
                              const float* __restrict__ logits_w,
                              const float* __restrict__ logits_b,
                              float* __restrict__ out) {
    int t = threadIdx.x;
    if (t >= BS * NMASK * NCLS) return;
    int b = t / (NMASK * NCLS);
    int r = t % (NMASK * NCLS);
    int m = r / NCLS, k = r % NCLS;
    float s = 0.0f;
    #pragma unroll
    for (int c = 0; c < CH; ++c)
        s += ws_pooled[b * (NMASK * CH) + m * CH + c] * logits_w[k * CH + c];
    out[b * (NMASK * NCLS) + r] = s * (1.0f / NF) + logits_b[k];
}

// ---------------------------------------------------------------------------
extern "C" void kernel_launch(void* const* d_in, const int* in_sizes, int n_in,
                              void* d_out, int out_size, void* d_ws, size_t ws_size,
                              hipStream_t stream) {
    const float* dec      = (const float*)d_in[0];
    const float* seg_w    = (const float*)d_in[1];
    const float* seg_b    = (const float*)d_in[2];
    const float* logits_w = (const float*)d_in[3];
    const float* logits_b = (const float*)d_in[4];

    float* out   = (float*)d_out;
    float* masks = out + BS * NMASK * NCLS;   // logits (48 floats) then masks
    float* ws    = (float*)d_ws;              // BS*NMASK*CH = 512 floats

    zero_ws_kernel <<<1, BS * NMASK * CH, 0, stream>>>(ws);
    masks_kernel   <<<dim3(HW / 256, BS), 256, 0, stream>>>(dec, seg_w, seg_b, masks);
    pooled_kernel  <<<dim3(BS * NF, HW / CHUNK), 256, 0, stream>>>(dec, masks, ws);
    logits_kernel  <<<1, 64, 0, stream>>>(ws, logits_w, logits_b, out);
}